// ModelNew_32607391711518
// MI455X (gfx1250) — compile-verified
//
#include <hip/hip_runtime.h>

// Fused 2x2x2 maxpool + bias + logsumexp(C) + relu for x[4,64,64,64,64] fp16.
// Data-movement bound (128 MiB in / 512 KiB out; HBM floor ~5.8us @ 23.3TB/s).
// Strategy: per-block 4D tile DMA (w,h,d + 4 channels) via the CDNA5 Tensor
// Data Mover into LDS, double buffered on TENSORcnt; pool + chunked online
// logsumexp from LDS with packed fp16 max.

typedef __attribute__((ext_vector_type(4))) unsigned int v4u;
typedef __attribute__((ext_vector_type(8))) int         v8i;
typedef __attribute__((ext_vector_type(4))) int         v4i;
typedef __attribute__((ext_vector_type(2))) _Float16    h2;   // packed half2

#define N_DIM 4
#define C_DIM 64
#define D_IN  64
#define H_IN  64
#define W_IN  64
#define D_OUT 32
#define H_OUT 32
#define W_OUT 32

#define CPB        4                    // channels per TDM chunk
#define NCHUNK     (C_DIM / CPB)        // 16 loop iterations
#define CH_HALF2   1024                 // half2 per channel slice (64w x 16h x 2d)
#define CHUNK_H2   (CPB * CH_HALF2)     // 4096 half2 = 16 KiB per chunk
#define CHUNK_B    (CHUNK_H2 * 4)       // bytes per chunk
#define TILE_HALF2 (2 * CHUNK_H2)       // double buffer: 32 KiB

// Issue one TDM 4D-tile load: tile 64(w) x 16(h) x 2(d) x 4(c), data_size=2B,
// strides 64 (h line), 4096 (d slice), 262144 (channel) elements, into LDS at
// byte offset lds_off. Descriptor packed per CDNA5 ISA 8.3-8.6.
// amdgpu-toolchain (clang-23) builtin arity: 6 args
//   (uint32x4 g0, int32x8 g1, int32x4 g2, int32x4 g3, int32x8 extra, i32 cpol)
__device__ __forceinline__ void tdm_load_chunk(unsigned long long ga, unsigned lds_off) {
  v4u g0;
  g0.x = 1u;                                              // count=1 (valid), user mode
  g0.y = lds_off;                                         // lds_addr [63:32]
  g0.z = (unsigned)(ga & 0xffffffffu);                    // global_addr lo
  g0.w = (unsigned)((ga >> 32) & 0x01ffffffu)             // global_addr[56:32]
         | (2u << 30);                                    // type=2 ("image")
  v8i g1;
  g1[0] = (int)(1u << 16);                                // data_size=1 -> 2 bytes
  g1[1] = (int)(64u << 16);                               // tensor_dim0 lo16 = 64
  g1[2] = (int)(64u << 16);                               // dim0 hi=0 | tensor_dim1 lo16 = 64
  g1[3] = (int)(64u << 16);                               // dim1 hi=0 | tile_dim0 = 64
  g1[4] = (int)(16u | (2u << 16));                        // tile_dim1=16 (h), tile_dim2=2 (d)
  g1[5] = 64;                                             // tensor_dim0_stride = 64 elems (h line)
  g1[6] = (int)(0x1000u << 16);                           // stride0 hi | tensor_dim1_stride lo16 = 4096
  g1[7] = 0;                                              // tensor_dim1_stride hi = 0
  v4i g2;
  g2[0] = 64;                                             // tensor_dim2 = 64 (d extent)
  g2[1] = 64;                                             // tensor_dim3 = 64 (c extent)
  g2[2] = (int)(D_IN * H_IN * W_IN);                      // tensor_dim2_stride = 262144 (c step)
  g2[3] = (int)(((unsigned)CPB) << 16);                   // stride hi=0 | tile_dim3 = 4
  v4i g3 = {0, 0, 0, 0};                                  // dim3 stride / dim4 unused
  v8i gx = {0, 0, 0, 0, 0, 0, 0, 0};                      // extra group (zero)
  __builtin_amdgcn_tensor_load_to_lds(g0, g1, g2, g3, gx, 0); // cpol = 0
}

struct SMem {
  h2    tiles[TILE_HALF2];   // double buffer, 32 KiB, at LDS offset 0
  float sbias[C_DIM];
};

__global__ __launch_bounds__(256)
void fused_pool_bias_lse_relu(const _Float16* __restrict__ x,
                              const _Float16* __restrict__ bias,
                              float* __restrict__ out) {
  // grid: (d_out=32, h_tile=4, n=4); block: 256 = 32 w_out x 8 h_out
  __shared__ __align__(16) SMem smem;

  const int d_out  = blockIdx.x;
  const int h_tile = blockIdx.y;
  const int n      = blockIdx.z;

  const int tid   = threadIdx.x;
  const int wo    = tid & 31;         // w_out lane
  const int h_loc = tid >> 5;         // 0..7 -> h_out = 8*h_tile + h_loc

  if (tid < C_DIM) smem.sbias[tid] = (float)bias[tid];

  // Byte address of tile start for channel 0 of this block.
  const unsigned long long baseElem =
      ((unsigned long long)(n * C_DIM) * D_IN + (unsigned long long)(2 * d_out)) *
          (unsigned long long)(H_IN * W_IN) +
      (unsigned long long)(16 * h_tile) * W_IN;
  const unsigned long long gbase = (unsigned long long)(uintptr_t)x + baseElem * 2ull;
  const unsigned long long chunkStrideB =
      (unsigned long long)CPB * (unsigned long long)(D_IN * H_IN * W_IN) * 2ull; // 2 MiB

  const bool wave0 = (tid < 32);      // uniform per wave32 -> scalar branch

  float m = -__builtin_inff();
  float s = 0.0f;

  if (wave0) tdm_load_chunk(gbase, 0u);

  // The TDM writes LDS behind the compiler's back (the descriptor carries the
  // LDS address as an integer). Escape the tile pointer into an asm with a
  // memory clobber so LLVM must assume the shared buffer gets written;
  // otherwise it folds the ds_loads to poison and deletes the compute.
  void* esc = (void*)&smem.tiles[0];
  asm volatile("" : "+v"(esc) : : "memory");

  const int h0 = 2 * h_loc;           // local h line within a channel slice

  for (int cb = 0; cb < NCHUNK; ++cb) {
    if (wave0) {
      if (cb + 1 < NCHUNK) {
        tdm_load_chunk(gbase + (unsigned long long)(cb + 1) * chunkStrideB,
                       (unsigned)((cb + 1) & 1) * CHUNK_B);
        __builtin_amdgcn_s_wait_tensorcnt(1);  // chunk cb's DMA done
      } else {
        __builtin_amdgcn_s_wait_tensorcnt(0);
      }
    }
    __syncthreads();                           // publish buf[cb&1] to all waves
    asm volatile("" : : : "memory");           // keep LDS loads below the barrier

    const h2* buf = &smem.tiles[(cb & 1) * CHUNK_H2];

    float v[CPB];
#pragma unroll
    for (int j = 0; j < CPB; ++j) {
      // channel slice j: half2 index = j*1024 + (d*16 + h)*32 + w_out
      const h2* b = buf + j * CH_HALF2;
      h2 a0 = b[(h0    ) * 32 + wo];           // d=0, h even
      h2 a1 = b[(h0 + 1) * 32 + wo];           // d=0, h odd
      h2 b0 = b[512 + (h0    ) * 32 + wo];     // d=1
      h2 b1 = b[512 + (h0 + 1) * 32 + wo];
      // 2x2x2 window max: packed fp16 max (exact) then widen to fp32.
      h2 mm = __builtin_elementwise_max(__builtin_elementwise_max(a0, a1),
                                        __builtin_elementwise_max(b0, b1));
      v[j] = fmaxf((float)mm.x, (float)mm.y) + smem.sbias[cb * CPB + j];
    }

    // Chunk-local logsumexp pieces (independent exps), then one (m,s) merge.
    float mc = fmaxf(fmaxf(v[0], v[1]), fmaxf(v[2], v[3]));
    float sc = __expf(v[0] - mc) + __expf(v[1] - mc) +
               __expf(v[2] - mc) + __expf(v[3] - mc);
    float mo = m;
    m = fmaxf(m, mc);                          // first chunk: m = mc
    s = s * __expf(mo - m) + sc * __expf(mc - m);  // exp(-inf)=0 handles init

    __syncthreads();                           // buf[cb&1] reusable for cb+2
  }

  float r = fmaxf(m + __logf(s), 0.0f);
  const int h_out = h_tile * 8 + h_loc;
  out[(((n * D_OUT) + d_out) * H_OUT + h_out) * W_OUT + wo] = r;
}

extern "C" void kernel_launch(void* const* d_in, const int* in_sizes, int n_in,
                              void* d_out, int out_size, void* d_ws, size_t ws_size,
                              hipStream_t stream) {
  const _Float16* x    = (const _Float16*)d_in[0];
  const _Float16* bias = (const _Float16*)d_in[1];
  float* out           = (float*)d_out;
  (void)in_sizes; (void)n_in; (void)out_size; (void)d_ws; (void)ws_size;

  dim3 grid(D_OUT, 4, N_DIM);   // 32 x 4 x 4 = 512 blocks
  dim3 block(256);              // 8 wave32
  hipLaunchKernelGGL(fused_pool_bias_lse_relu, grid, block, 0, stream, x, bias, out);
}